// H2OEvictor_86457691669172
// MI455X (gfx1250) — compile-verified
//
#include <hip/hip_runtime.h>
#include <math.h>

typedef __attribute__((ext_vector_type(2))) float v2f;
typedef __attribute__((ext_vector_type(4))) float v4f;
typedef __attribute__((ext_vector_type(8))) float v8f;

#define B_    2
#define H_    32
#define QN    4
#define S_    8192
#define D_    128
#define CHUNK 1024
#define NCH   (S_/CHUNK)   // 8
#define KEEP  4096
#define KSEL  4092
#define SINK  4

// Staged-tile geometry for async global->LDS streaming
#define TKEYS   128                 // keys per staged tile
#define NST     (CHUNK/TKEYS)       // 8 stages per chunk
#define TROW    132                 // padded row stride (floats) -> no LDS bank conflicts
#define SCROW   1028                // padded score row stride (floats)
#define SMEM_FLOATS (QN*SCROW + 256 + 2*TKEYS*TROW)
#define SMEM_BYTES  (SMEM_FLOATS*4)

// Issue one staged tile (TKEYS x 128 floats) as async global->LDS b128 copies.
// 256 threads * 16 chunks * 16B = 512 KB/8 = 64 KB tile. ASYNCcnt += 16 per thread.
__device__ __forceinline__ void stage_issue(const float* __restrict__ gbase,
                                            float* __restrict__ lbase, int tid)
{
  #pragma unroll
  for (int i = 0; i < 16; ++i) {
    const int cidx = tid + 256 * i;        // 4096 16B-chunks per tile
    const int row  = cidx >> 5;            // 32 chunks per 512B row
    const int c4   = cidx & 31;
    const float* g = gbase + (long)row * D_ + c4 * 4;
    float*       l = lbase + row * TROW + c4 * 4;
    const unsigned      loff = (unsigned)(unsigned long long)l;
    const unsigned long long ga = (unsigned long long)g;
    asm volatile("global_load_async_to_lds_b128 %0, %1, off"
                 :: "v"(loff), "v"(ga) : "memory");
  }
}

__device__ __forceinline__ void wait_async_16()
{ asm volatile("s_wait_asynccnt 0x10" ::: "memory"); }
__device__ __forceinline__ void wait_async_0()
{ asm volatile("s_wait_asynccnt 0x0" ::: "memory"); }

// ---------------------------------------------------------------------------
// Kernel 1: per (head, chunk) flash attention partials using fp32 WMMA, with
// double-buffered async global->LDS staging of K and V tiles.
// ---------------------------------------------------------------------------
__global__ __launch_bounds__(256) void attn_chunk_kernel(
    const float* __restrict__ qin, const float* __restrict__ kin,
    const float* __restrict__ vin,
    float* __restrict__ u_ws, float* __restrict__ opart,
    float* __restrict__ m_ws, float* __restrict__ l_ws)
{
  extern __shared__ float smem[];
  float* s_sc = smem;                        // QN * SCROW  (scores, then u)
  float* rbuf = smem + QN * SCROW;           // 256
  float* buf0 = rbuf + 256;                  // TKEYS * TROW
  float* buf1 = buf0 + TKEYS * TROW;         // TKEYS * TROW

  const int blk   = blockIdx.x;
  const int head  = blk / NCH;               // b*H + h
  const int chunk = blk % NCH;
  const int tid   = threadIdx.x;
  const int lane  = tid & 31;
  const int wave  = tid >> 5;                // 8 waves
  const int m     = lane & 15;
  const int hi    = lane >> 4;
  const float scale = 0.08838834764831845f;  // 1/sqrt(128)

  const float* Qp = qin + (long)head * QN * D_;
  const float* Kp = kin + ((long)head * S_ + (long)chunk * CHUNK) * D_;
  const float* Vp = vin + ((long)head * S_ + (long)chunk * CHUNK) * D_;

  // Kick off K stage 0 before anything else.
  stage_issue(Kp, buf0, tid);

  // ---- A fragments: Q rows 0..3, zero-padded to 16 rows.
  v2f aq[32];
  #pragma unroll
  for (int c = 0; c < 32; ++c) {
    v2f a;
    if (m < QN) a = *(const v2f*)(Qp + m * D_ + 4 * c + 2 * hi);
    else        { a.x = 0.f; a.y = 0.f; }
    aq[c] = a;
  }

  // ---- Phase 1: scores. Each stage holds 128 keys; wave w owns keys
  // [w*16, w*16+16) of the stage -> 32 WMMA per wave per stage.
  for (int t = 0; t < NST; ++t) {
    if (t + 1 < NST) {
      stage_issue(Kp + (long)(t + 1) * TKEYS * D_, (t & 1) ? buf0 : buf1, tid);
      wait_async_16();                 // stage t complete (in-order per wave)
    } else {
      wait_async_0();
    }
    __syncthreads();                   // all waves' copies visible

    const float* kb = (t & 1) ? buf1 : buf0;
    v8f acc = {};
    #pragma unroll
    for (int c = 0; c < 32; ++c) {
      v2f bk = *(const v2f*)(kb + (wave * 16 + m) * TROW + 4 * c + 2 * hi);
      acc = __builtin_amdgcn_wmma_f32_16x16x4_f32(false, aq[c], false, bk,
                                                  (short)0, acc, false, false);
    }
    if (hi == 0) {
      #pragma unroll
      for (int r = 0; r < QN; ++r)
        s_sc[r * SCROW + t * TKEYS + wave * 16 + m] = acc[r] * scale;
    }
    __syncthreads();                   // done with buf[t&1] before reuse
  }

  // Prefetch V stage 0 now; it completes under the softmax phase.
  stage_issue(Vp, buf0, tid);

  // ---- Phase 2: per-q chunk max / exp / sum; u -> global workspace
  float mqv[QN], lqv[QN];
  for (int qq = 0; qq < QN; ++qq) {
    float mx = -1e30f;
    for (int j = tid; j < CHUNK; j += 256) mx = fmaxf(mx, s_sc[qq * SCROW + j]);
    rbuf[tid] = mx; __syncthreads();
    for (int off = 128; off > 0; off >>= 1) {
      if (tid < off) rbuf[tid] = fmaxf(rbuf[tid], rbuf[tid + off]);
      __syncthreads();
    }
    const float mq = rbuf[0]; __syncthreads();

    float sm = 0.f;
    for (int j = tid; j < CHUNK; j += 256) {
      float e = __expf(s_sc[qq * SCROW + j] - mq);
      s_sc[qq * SCROW + j] = e;
      u_ws[((long)head * QN + qq) * S_ + (long)chunk * CHUNK + j] = e;
      sm += e;
    }
    rbuf[tid] = sm; __syncthreads();
    for (int off = 128; off > 0; off >>= 1) {
      if (tid < off) rbuf[tid] += rbuf[tid + off];
      __syncthreads();
    }
    mqv[qq] = mq; lqv[qq] = rbuf[0]; __syncthreads();
  }
  if (tid == 0) {
    for (int qq = 0; qq < QN; ++qq) {
      m_ws[(head * NCH + chunk) * QN + qq] = mqv[qq];
      l_ws[(head * NCH + chunk) * QN + qq] = lqv[qq];
    }
  }

  // ---- Phase 3: O_part = u * V via WMMA; wave owns d-tile [wave*16, +16).
  const int dbase = wave * 16;
  v8f vacc = {};
  for (int t = 0; t < NST; ++t) {
    if (t + 1 < NST) {
      stage_issue(Vp + (long)(t + 1) * TKEYS * D_, (t & 1) ? buf0 : buf1, tid);
      wait_async_16();
    } else {
      wait_async_0();
    }
    __syncthreads();

    const float* vb = (t & 1) ? buf1 : buf0;
    #pragma unroll
    for (int kc = 0; kc < TKEYS / 4; ++kc) {
      const int kloc = kc * 4;
      v2f au;
      if (m < QN) {
        au.x = s_sc[m * SCROW + t * TKEYS + kloc + 2 * hi];
        au.y = s_sc[m * SCROW + t * TKEYS + kloc + 2 * hi + 1];
      } else { au.x = 0.f; au.y = 0.f; }
      v2f bv;
      bv.x = vb[(kloc + 2 * hi)     * TROW + dbase + m];
      bv.y = vb[(kloc + 2 * hi + 1) * TROW + dbase + m];
      vacc = __builtin_amdgcn_wmma_f32_16x16x4_f32(false, au, false, bv,
                                                   (short)0, vacc, false, false);
    }
    __syncthreads();
  }
  if (hi == 0) {
    #pragma unroll
    for (int r = 0; r < QN; ++r)
      opart[(((long)head * NCH + chunk) * QN + r) * D_ + dbase + m] = vacc[r];
  }
}

// ---------------------------------------------------------------------------
// Kernel 2: combine chunk partials per head -> attn_out, chunk scales
// ---------------------------------------------------------------------------
__global__ __launch_bounds__(128) void combine_kernel(
    const float* __restrict__ opart, const float* __restrict__ m_ws,
    const float* __restrict__ l_ws,
    float* __restrict__ scale_ws, float* __restrict__ attn_out)
{
  __shared__ float scl[NCH][QN];
  const int head = blockIdx.x;
  const int tid  = threadIdx.x;
  if (tid < QN) {
    const int qq = tid;
    float mf = -1e30f;
    for (int c = 0; c < NCH; ++c)
      mf = fmaxf(mf, m_ws[(head * NCH + c) * QN + qq]);
    float lf = 0.f;
    for (int c = 0; c < NCH; ++c)
      lf += l_ws[(head * NCH + c) * QN + qq] *
            __expf(m_ws[(head * NCH + c) * QN + qq] - mf);
    const float inv = 1.f / lf;
    for (int c = 0; c < NCH; ++c) {
      float sc = __expf(m_ws[(head * NCH + c) * QN + qq] - mf) * inv;
      scl[c][qq] = sc;
      scale_ws[(head * NCH + c) * QN + qq] = sc;
    }
  }
  __syncthreads();
  for (int idx = tid; idx < QN * D_; idx += 128) {
    const int qq = idx / D_;
    const int d  = idx % D_;
    float o = 0.f;
    for (int c = 0; c < NCH; ++c)
      o += scl[c][qq] * opart[(((long)head * NCH + c) * QN + qq) * D_ + d];
    attn_out[((long)head * QN + qq) * D_ + d] = o;
  }
}

// ---------------------------------------------------------------------------
// Kernel 3: head_avg[b,s] = (1/H) * sum_{h,q} attn[b,h,q,s]
// ---------------------------------------------------------------------------
__global__ __launch_bounds__(256) void headavg_kernel(
    const float* __restrict__ u_ws, const float* __restrict__ scale_ws,
    float* __restrict__ head_avg)
{
  const int tid   = threadIdx.x;
  const int blk   = blockIdx.x;
  const int b     = blk / (S_ / 256);
  const int s     = (blk % (S_ / 256)) * 256 + tid;
  const int chunk = s / CHUNK;
  float a = 0.f;
  for (int h = 0; h < H_; ++h) {
    const int head = b * H_ + h;
    for (int qq = 0; qq < QN; ++qq)
      a += u_ws[((long)head * QN + qq) * S_ + s] *
           scale_ws[(head * NCH + chunk) * QN + qq];
  }
  head_avg[b * S_ + s] = a * (1.f / H_);
}

// ---------------------------------------------------------------------------
// Kernel 4: MSB-first 8-bit radix select for the 4092-nd largest candidate.
// ---------------------------------------------------------------------------
__global__ __launch_bounds__(256) void select_kernel(
    const float* __restrict__ head_avg, unsigned int* __restrict__ sel)
{
  __shared__ unsigned int hist[256];
  __shared__ unsigned int sh_prefix, sh_rem;
  const int b   = blockIdx.x;
  const int tid = threadIdx.x;
  unsigned int prefix = 0;
  unsigned int rem    = KSEL;
  for (int round = 0; round < 4; ++round) {
    const int shift = 24 - 8 * round;
    hist[tid] = 0; __syncthreads();
    for (int j = SINK + tid; j < S_; j += 256) {
      unsigned int bits = __float_as_uint(head_avg[b * S_ + j]);
      bool match = (round == 0) ||
                   ((bits >> (shift + 8)) == (prefix >> (shift + 8)));
      if (match) atomicAdd(&hist[(bits >> shift) & 255], 1u);
    }
    __syncthreads();
    if (tid == 0) {
      unsigned int cum = 0, chosen = 0;
      for (int bkt = 255; bkt >= 0; --bkt) {
        unsigned int hc = hist[bkt];
        if (cum + hc >= rem) { chosen = (unsigned int)bkt; break; }
        cum += hc;
      }
      sh_prefix = prefix | (chosen << shift);
      sh_rem    = rem - cum;
    }
    __syncthreads();
    prefix = sh_prefix; rem = sh_rem;
    __syncthreads();
  }
  if (tid == 0) { sel[b * 2] = prefix; sel[b * 2 + 1] = rem; }
}

// ---------------------------------------------------------------------------
// Kernel 5: stable compaction -> keep_idx (ascending token order).
// ---------------------------------------------------------------------------
__global__ __launch_bounds__(1024) void keep_kernel(
    const float* __restrict__ head_avg, const unsigned int* __restrict__ sel,
    int* __restrict__ keep_idx)
{
  __shared__ unsigned int sg[1024], se[1024];
  const int b   = blockIdx.x;
  const int tid = threadIdx.x;
  const unsigned int T    = sel[b * 2];
  const unsigned int ties = sel[b * 2 + 1];
  const int s0 = tid * 8;
  unsigned int cg = 0, ce = 0, fg = 0, fe = 0;
  for (int i = 0; i < 8; ++i) {
    const int s = s0 + i;
    const unsigned int bits = __float_as_uint(head_avg[b * S_ + s]);
    const bool sink = s < SINK;
    const bool gt   = sink || (bits > T);
    const bool eq   = (!sink) && (bits == T);
    if (gt)      { cg++; fg |= (1u << i); }
    else if (eq) { ce++; fe |= (1u << i); }
  }
  sg[tid] = cg; se[tid] = ce; __syncthreads();
  for (int off = 1; off < 1024; off <<= 1) {
    unsigned int ag = 0, ae = 0;
    if (tid >= off) { ag = sg[tid - off]; ae = se[tid - off]; }
    __syncthreads();
    sg[tid] += ag; se[tid] += ae;
    __syncthreads();
  }
  unsigned int g_off = sg[tid] - cg;
  unsigned int e_off = se[tid] - ce;
  for (int i = 0; i < 8; ++i) {
    const int s = s0 + i;
    if (fg & (1u << i)) {
      const unsigned int pos = g_off + (e_off < ties ? e_off : ties);
      keep_idx[b * KEEP + pos] = s;
      g_off++;
    } else if (fe & (1u << i)) {
      if (e_off < ties) keep_idx[b * KEEP + g_off + e_off] = s;
      e_off++;
    }
  }
}

// ---------------------------------------------------------------------------
// Kernel 6: gather kept K/V rows (one wave per row, b128 traffic)
// ---------------------------------------------------------------------------
__global__ __launch_bounds__(256) void gather_kernel(
    const float* __restrict__ kin, const float* __restrict__ vin,
    const int* __restrict__ keep_idx,
    float* __restrict__ new_key, float* __restrict__ new_value)
{
  const int tid  = threadIdx.x;
  const int row  = blockIdx.x * 8 + (tid >> 5);
  const int lane = tid & 31;
  const int b    = row / (H_ * KEEP);
  const int rem  = row % (H_ * KEEP);
  const int h    = rem / KEEP;
  const int j    = rem % KEEP;
  const int src  = keep_idx[b * KEEP + j];
  const long srcoff = (((long)(b * H_ + h)) * S_ + src) * D_;
  const long dstoff = (((long)(b * H_ + h)) * KEEP + j) * D_;
  const v4f* ks = (const v4f*)(kin + srcoff);
  const v4f* vs = (const v4f*)(vin + srcoff);
  v4f* kd = (v4f*)(new_key + dstoff);
  v4f* vd = (v4f*)(new_value + dstoff);
  kd[lane] = ks[lane];
  vd[lane] = vs[lane];
}

// ---------------------------------------------------------------------------
extern "C" void kernel_launch(void* const* d_in, const int* in_sizes, int n_in,
                              void* d_out, int out_size, void* d_ws, size_t ws_size,
                              hipStream_t stream)
{
  (void)in_sizes; (void)n_in; (void)out_size; (void)ws_size;
  const float* q  = (const float*)d_in[0];
  const float* kk = (const float*)d_in[1];
  const float* vv = (const float*)d_in[2];

  float* out      = (float*)d_out;
  float* attn_out = out;                                   // B*H*Q*D = 32768
  float* new_key  = out + (long)B_ * H_ * QN * D_;
  float* new_val  = new_key + (long)B_ * H_ * KEEP * D_;

  float* ws = (float*)d_ws;
  long o = 0;
  float* u_ws     = ws + o; o += (long)B_ * H_ * QN * S_;        // 2M floats
  float* opart    = ws + o; o += (long)B_ * H_ * NCH * QN * D_;  // 256K
  float* m_ws     = ws + o; o += B_ * H_ * NCH * QN;
  float* l_ws     = ws + o; o += B_ * H_ * NCH * QN;
  float* scale_ws = ws + o; o += B_ * H_ * NCH * QN;
  float* head_avg = ws + o; o += B_ * S_;
  unsigned int* sel = (unsigned int*)(ws + o); o += 4;
  int* keep_idx   = (int*)(ws + o);

  attn_chunk_kernel<<<B_ * H_ * NCH, 256, SMEM_BYTES, stream>>>(
      q, kk, vv, u_ws, opart, m_ws, l_ws);
  combine_kernel<<<B_ * H_, 128, 0, stream>>>(opart, m_ws, l_ws, scale_ws,
                                              attn_out);
  headavg_kernel<<<B_ * (S_ / 256), 256, 0, stream>>>(u_ws, scale_ws, head_avg);
  select_kernel<<<B_, 256, 0, stream>>>(head_avg, sel);
  keep_kernel<<<B_, 1024, 0, stream>>>(head_avg, sel, keep_idx);
  gather_kernel<<<(B_ * H_ * KEEP) / 8, 256, 0, stream>>>(kk, vv, keep_idx,
                                                          new_key, new_val);
}